// ViT_VDP_85255100826172
// MI455X (gfx1250) — compile-verified
//
#include <hip/hip_runtime.h>

typedef __attribute__((ext_vector_type(2))) float v2f;
typedef __attribute__((ext_vector_type(8))) float v8f;

#define VAR_INIT 1e-8f
#define LN_EPS   1e-5f
#define B_   8
#define S_   196
#define K_   1024
#define H_   16
#define DH_  64
#define M_   (B_ * S_)
#define NL_  7

static __device__ __forceinline__ v8f wmma_f32(v2f a, v2f b, v8f c) {
  // D = A(16x4 f32) * B(4x16 f32) + C(16x16 f32)
  return __builtin_amdgcn_wmma_f32_16x16x4_f32(false, a, false, b, (short)0, c,
                                               false, false);
}

// ---------------------------------------------------------------------------
// Patchify + embedding linear_vdp (vx==0 on input) + positional encoding.
// v = VAR_INIT * sum(p^2) + VAR_INIT  (same for all output cols of a row)
// ---------------------------------------------------------------------------
__global__ __launch_bounds__(256) void patch_embed_k(
    const float* __restrict__ x, const float* __restrict__ Wemb,
    const float* __restrict__ bemb, float* __restrict__ h,
    float* __restrict__ vx) {
  int m = blockIdx.x;                        // token row 0..1567
  int n = blockIdx.y * 256 + threadIdx.x;    // output col 0..1023
  int b = m / S_;
  int t = m % S_;
  int py = t / 14, px = t % 14;
  __shared__ float xs[256];
  __shared__ float red[256];
  int i = threadIdx.x;
  int r = i >> 4, c = i & 15;
  float pv = x[((size_t)b * 224 + py * 16 + r) * 224 + px * 16 + c];
  xs[i] = pv;
  red[i] = pv * pv;
  __syncthreads();
  for (int s = 128; s > 0; s >>= 1) {
    if (i < s) red[i] += red[i + s];
    __syncthreads();
  }
  float s2 = red[0];
  const float* wrow = Wemb + (size_t)n * 256;
  float acc = 0.f;
#pragma unroll 8
  for (int j = 0; j < 256; ++j) acc += xs[j] * wrow[j];
  float fe = (float)(2 * (n >> 1));
  float ang = (float)t * __expf(-logf(10000.0f) * fe / (float)K_);
  float pe = (n & 1) ? __cosf(ang) : __sinf(ang);
  h[(size_t)m * K_ + n] = acc + bemb[n] + pe;
  vx[(size_t)m * K_ + n] = VAR_INIT * s2 + VAR_INIT;
}

// ---------------------------------------------------------------------------
// rs[row] = VAR_INIT * sum_k(x^2 + vx)  (generic Kdim; also used for head)
// ---------------------------------------------------------------------------
__global__ __launch_bounds__(256) void rowsum_k(const float* __restrict__ X,
                                                const float* __restrict__ VX,
                                                float* __restrict__ rs,
                                                int Kdim) {
  int m = blockIdx.x;
  __shared__ float red[256];
  float a = 0.f;
  for (int j = threadIdx.x; j < Kdim; j += 256) {
    float xv = X[(size_t)m * Kdim + j];
    float vv = VX[(size_t)m * Kdim + j];
    a += xv * xv + vv;
  }
  red[threadIdx.x] = a;
  __syncthreads();
  for (int s = 128; s > 0; s >>= 1) {
    if (threadIdx.x < s) red[threadIdx.x] += red[threadIdx.x + s];
    __syncthreads();
  }
  if (threadIdx.x == 0) rs[m] = VAR_INIT * red[0];
}

// ---------------------------------------------------------------------------
// layernorm_vdp, in place, one block per row of 1024
// ---------------------------------------------------------------------------
__global__ __launch_bounds__(256) void layernorm_vdp_k(
    float* __restrict__ X, float* __restrict__ VX, const float* __restrict__ g,
    const float* __restrict__ bb) {
  int m = blockIdx.x;
  int t = threadIdx.x;
  __shared__ float red[256];
  float xs[4], vs[4];
  float s = 0.f;
#pragma unroll
  for (int u = 0; u < 4; ++u) {
    int n = t + u * 256;
    xs[u] = X[(size_t)m * K_ + n];
    vs[u] = VX[(size_t)m * K_ + n];
    s += xs[u];
  }
  red[t] = s;
  __syncthreads();
  for (int st = 128; st > 0; st >>= 1) {
    if (t < st) red[t] += red[t + st];
    __syncthreads();
  }
  float mu = red[0] * (1.f / K_);
  __syncthreads();
  float d = 0.f;
#pragma unroll
  for (int u = 0; u < 4; ++u) {
    float dd = xs[u] - mu;
    d += dd * dd;
  }
  red[t] = d;
  __syncthreads();
  for (int st = 128; st > 0; st >>= 1) {
    if (t < st) red[t] += red[t + st];
    __syncthreads();
  }
  float var = red[0] * (1.f / K_);
  float rstd = rsqrtf(var + LN_EPS);
#pragma unroll
  for (int u = 0; u < 4; ++u) {
    int n = t + u * 256;
    float xn = (xs[u] - mu) * rstd;
    float gg = g[n];
    float grs = gg * rstd;
    X[(size_t)m * K_ + n] = gg * xn + bb[n];
    VX[(size_t)m * K_ + n] = grs * grs * vs[u] + VAR_INIT * xn * xn + VAR_INIT;
  }
}

// ---------------------------------------------------------------------------
// WMMA linear_vdp GEMM:  OM = X @ W^T (+b) ; OV = VX @ (W*W)^T + rs[m] (+eps)
// flags: 1=bias, 2=relu_vdp, 4=add residual (RM,RV)
// Block: 256 threads = 8 waves over a 64(M) x 32(N) tile; K chunk 32 in LDS.
// ---------------------------------------------------------------------------
#define GTM 64
#define GTN 32
#define GTK 32
__global__ __launch_bounds__(256) void gemm_vdp_k(
    const float* __restrict__ X, const float* __restrict__ VX,
    const float* __restrict__ W, const float* __restrict__ bias,
    const float* __restrict__ rs, const float* __restrict__ RM,
    const float* __restrict__ RV, float* __restrict__ OM,
    float* __restrict__ OV, int Mdim, int Ndim, int Kdim, int flags) {
  __shared__ float Xs[GTM][GTK + 1];
  __shared__ float VXs[GTM][GTK + 1];
  __shared__ float Ws[GTN][GTK + 1];
  __shared__ float W2s[GTN][GTK + 1];
  int tid = threadIdx.x;
  int lane = tid & 31;
  int wave = tid >> 5;  // 0..7
  int wm = wave >> 1;   // 0..3 (M sub-tile)
  int wn = wave & 1;    // 0..1 (N sub-tile)
  int row0 = blockIdx.y * GTM;
  int col0 = blockIdx.x * GTN;
  v8f accM = {0, 0, 0, 0, 0, 0, 0, 0};
  v8f accV = {0, 0, 0, 0, 0, 0, 0, 0};
  int kc = tid & (GTK - 1);
  int rbase = tid >> 5;  // 8 rows per pass
  int ml = lane & 15;
  int khalf = (lane >> 4) << 1;
  for (int k0 = 0; k0 < Kdim; k0 += GTK) {
    for (int p = 0; p < GTM; p += 8) {
      int rr = rbase + p;
      int gm = row0 + rr;
      if (gm >= Mdim) gm = Mdim - 1;
      Xs[rr][kc] = X[(size_t)gm * Kdim + k0 + kc];
      VXs[rr][kc] = VX[(size_t)gm * Kdim + k0 + kc];
    }
    for (int p = 0; p < GTN; p += 8) {
      int rr = rbase + p;
      float wv = W[(size_t)(col0 + rr) * Kdim + k0 + kc];
      Ws[rr][kc] = wv;
      W2s[rr][kc] = wv * wv;
    }
    __syncthreads();
#pragma unroll
    for (int kk = 0; kk < GTK; kk += 4) {
      int ko = kk + khalf;
      v2f a, av, b, b2;
      a.x = Xs[wm * 16 + ml][ko];
      a.y = Xs[wm * 16 + ml][ko + 1];
      av.x = VXs[wm * 16 + ml][ko];
      av.y = VXs[wm * 16 + ml][ko + 1];
      b.x = Ws[wn * 16 + ml][ko];
      b.y = Ws[wn * 16 + ml][ko + 1];
      b2.x = W2s[wn * 16 + ml][ko];
      b2.y = W2s[wn * 16 + ml][ko + 1];
      accM = wmma_f32(a, b, accM);
      accV = wmma_f32(av, b2, accV);
    }
    __syncthreads();
  }
  int nl = lane & 15;
  int mh = (lane >> 4) * 8;
  int n = col0 + wn * 16 + nl;
  bool hasb = (flags & 1) != 0;
  bool dorelu = (flags & 2) != 0;
  bool resid = (flags & 4) != 0;
  float bn = hasb ? bias[n] : 0.f;
#pragma unroll
  for (int r = 0; r < 8; ++r) {
    int m = row0 + wm * 16 + mh + r;
    if (m < Mdim) {
      float mm = accM[r] + bn;
      float vv = accV[r] + rs[m] + (hasb ? VAR_INIT : 0.f);
      if (dorelu) {
        vv = (mm > 0.f) ? vv : 0.f;
        mm = fmaxf(mm, 0.f);
      }
      if (resid) {
        mm += RM[(size_t)m * Ndim + n];
        vv += RV[(size_t)m * Ndim + n];
      }
      OM[(size_t)m * Ndim + n] = mm;
      OV[(size_t)m * Ndim + n] = vv;
    }
  }
}

// ---------------------------------------------------------------------------
// Attention scores: a = q@k^T / 32 ; va = (vq@(vk+k^2) + q^2@vk) / 1024
// One wave per 16x16 tile per (b,h).
// ---------------------------------------------------------------------------
__global__ __launch_bounds__(32) void attn_scores_k(
    const float* __restrict__ Q, const float* __restrict__ VQ,
    const float* __restrict__ Kmat, const float* __restrict__ VK,
    float* __restrict__ A, float* __restrict__ VA) {
  int bh = blockIdx.z;
  int b = bh / H_, hh = bh % H_;
  int i0 = blockIdx.y * 16, j0 = blockIdx.x * 16;
  int lane = threadIdx.x;
  int ml = lane & 15;
  int khalf = (lane >> 4) << 1;
  int qi = i0 + ml;
  if (qi >= S_) qi = S_ - 1;
  int kj = j0 + ml;
  if (kj >= S_) kj = S_ - 1;
  const float* qrow = Q + ((size_t)b * S_ + qi) * K_ + hh * DH_;
  const float* vqrow = VQ + ((size_t)b * S_ + qi) * K_ + hh * DH_;
  const float* krow = Kmat + ((size_t)b * S_ + kj) * K_ + hh * DH_;
  const float* vkrow = VK + ((size_t)b * S_ + kj) * K_ + hh * DH_;
  v8f accM = {0, 0, 0, 0, 0, 0, 0, 0};
  v8f accV = {0, 0, 0, 0, 0, 0, 0, 0};
#pragma unroll 4
  for (int kk = 0; kk < DH_; kk += 4) {
    int ko = kk + khalf;
    v2f a = *(const v2f*)(qrow + ko);
    v2f av = *(const v2f*)(vqrow + ko);
    v2f bk = *(const v2f*)(krow + ko);
    v2f bvk = *(const v2f*)(vkrow + ko);
    v2f a2;
    a2.x = a.x * a.x;
    a2.y = a.y * a.y;
    v2f b2;
    b2.x = bvk.x + bk.x * bk.x;
    b2.y = bvk.y + bk.y * bk.y;
    accM = wmma_f32(a, bk, accM);
    accV = wmma_f32(av, b2, accV);
    accV = wmma_f32(a2, bvk, accV);
  }
  int nl = lane & 15;
  int mh = (lane >> 4) * 8;
  const float invrd = 1.f / 32.f;
  float* Abh = A + (size_t)bh * S_ * S_;
  float* VAbh = VA + (size_t)bh * S_ * S_;
#pragma unroll
  for (int r = 0; r < 8; ++r) {
    int i = i0 + mh + r, j = j0 + nl;
    if (i < S_ && j < S_) {
      Abh[(size_t)i * S_ + j] = accM[r] * invrd;
      VAbh[(size_t)i * S_ + j] = accV[r] * (invrd * invrd);
    }
  }
}

// ---------------------------------------------------------------------------
// softmax_vdp over rows of length 196 (in place)
// ---------------------------------------------------------------------------
__global__ __launch_bounds__(256) void softmax_vdp_k(float* __restrict__ A,
                                                     float* __restrict__ VA) {
  size_t row = blockIdx.x;
  float* a = A + row * S_;
  float* va = VA + row * S_;
  __shared__ float red[256];
  int t = threadIdx.x;
  float x = (t < S_) ? a[t] : -3.0e38f;
  red[t] = x;
  __syncthreads();
  for (int s = 128; s > 0; s >>= 1) {
    if (t < s) red[t] = fmaxf(red[t], red[t + s]);
    __syncthreads();
  }
  float mx = red[0];
  __syncthreads();
  float e = (t < S_) ? __expf(x - mx) : 0.f;
  red[t] = e;
  __syncthreads();
  for (int s = 128; s > 0; s >>= 1) {
    if (t < s) red[t] += red[t + s];
    __syncthreads();
  }
  float inv = 1.f / red[0];
  if (t < S_) {
    float p = e * inv;
    float j = p * (1.f - p);
    a[t] = p;
    va[t] = j * j * va[t];
  }
}

// ---------------------------------------------------------------------------
// Attention output quadratic_vdp + identity residual:
//   o = p@v ; vo = vp@(vv+v^2) + p^2@vv ; OM = RM + o ; OV = RV + vo
// One wave per 16(M) x 16(N=d) tile per (b,h). K loop over 196 (mult of 4).
// ---------------------------------------------------------------------------
__global__ __launch_bounds__(32) void attn_out_k(
    const float* __restrict__ P, const float* __restrict__ VP,
    const float* __restrict__ Vv, const float* __restrict__ VVv,
    const float* __restrict__ RM, const float* __restrict__ RV,
    float* __restrict__ OM, float* __restrict__ OV) {
  int bh = blockIdx.z;
  int b = bh / H_, hh = bh % H_;
  int i0 = blockIdx.y * 16, d0 = blockIdx.x * 16;
  int lane = threadIdx.x;
  int ml = lane & 15;
  int khalf = (lane >> 4) << 1;
  int pi = i0 + ml;
  if (pi >= S_) pi = S_ - 1;
  const float* prow = P + ((size_t)bh * S_ + pi) * S_;
  const float* vprow = VP + ((size_t)bh * S_ + pi) * S_;
  int nd = d0 + ml;  // per-lane N column for B fragments
  const float* vbase = Vv + (size_t)b * S_ * K_ + hh * DH_ + nd;
  const float* wbase = VVv + (size_t)b * S_ * K_ + hh * DH_ + nd;
  v8f accM = {0, 0, 0, 0, 0, 0, 0, 0};
  v8f accV = {0, 0, 0, 0, 0, 0, 0, 0};
#pragma unroll 4
  for (int kk = 0; kk < S_; kk += 4) {
    int ko = kk + khalf;
    v2f a = *(const v2f*)(prow + ko);
    v2f av = *(const v2f*)(vprow + ko);
    v2f a2;
    a2.x = a.x * a.x;
    a2.y = a.y * a.y;
    float vj0 = vbase[(size_t)ko * K_];
    float vj1 = vbase[(size_t)(ko + 1) * K_];
    float wj0 = wbase[(size_t)ko * K_];
    float wj1 = wbase[(size_t)(ko + 1) * K_];
    v2f bb = {vj0, vj1};
    v2f b2 = {wj0 + vj0 * vj0, wj1 + vj1 * vj1};
    v2f b3 = {wj0, wj1};
    accM = wmma_f32(a, bb, accM);
    accV = wmma_f32(av, b2, accV);
    accV = wmma_f32(a2, b3, accV);
  }
  int nl = lane & 15;
  int mh = (lane >> 4) * 8;
#pragma unroll
  for (int r = 0; r < 8; ++r) {
    int i = i0 + mh + r;
    int d = d0 + nl;
    if (i < S_) {
      size_t idx = ((size_t)b * S_ + i) * K_ + hh * DH_ + d;
      OM[idx] = RM[idx] + accM[r];
      OV[idx] = RV[idx] + accV[r];
    }
  }
}

// ---------------------------------------------------------------------------
// Mean pool over tokens; variance additionally / S_ (reference: /NPAT^2)
// ---------------------------------------------------------------------------
__global__ __launch_bounds__(256) void pool_k(const float* __restrict__ Hh,
                                              const float* __restrict__ VX,
                                              float* __restrict__ hp,
                                              float* __restrict__ vp) {
  int bq = blockIdx.x;
  int n = blockIdx.y * 256 + threadIdx.x;
  float sm = 0.f, sv = 0.f;
  for (int s = 0; s < S_; ++s) {
    sm += Hh[((size_t)bq * S_ + s) * K_ + n];
    sv += VX[((size_t)bq * S_ + s) * K_ + n];
  }
  hp[(size_t)bq * K_ + n] = sm / (float)S_;
  vp[(size_t)bq * K_ + n] = sv / ((float)S_ * (float)S_);
}

// ---------------------------------------------------------------------------
// Small dense linear_vdp for the classifier head (8 rows only)
// ---------------------------------------------------------------------------
__global__ __launch_bounds__(256) void head_linear_k(
    const float* __restrict__ Xp, const float* __restrict__ VXp,
    const float* __restrict__ W, const float* __restrict__ bias,
    const float* __restrict__ rsb, float* __restrict__ OM,
    float* __restrict__ OV, int Ndim, int Kdim, int dorelu) {
  int bq = blockIdx.x;
  int n = blockIdx.y * 256 + threadIdx.x;
  if (n >= Ndim) return;
  const float* xr = Xp + (size_t)bq * Kdim;
  const float* vr = VXp + (size_t)bq * Kdim;
  const float* wr = W + (size_t)n * Kdim;
  float am = 0.f, av = 0.f;
  for (int j = 0; j < Kdim; ++j) {
    float w = wr[j];
    am += xr[j] * w;
    av += vr[j] * w * w;
  }
  am += bias[n];
  av += rsb[bq] + VAR_INIT;
  if (dorelu) {
    av = (am > 0.f) ? av : 0.f;
    am = fmaxf(am, 0.f);
  }
  OM[(size_t)bq * Ndim + n] = am;
  OV[(size_t)bq * Ndim + n] = av;
}

// ---------------------------------------------------------------------------
// Final softmax_vdp over 1000 logits -> d_out (means then variances)
// ---------------------------------------------------------------------------
__global__ __launch_bounds__(256) void softmax_out_k(
    const float* __restrict__ LM, const float* __restrict__ LV,
    float* __restrict__ outM, float* __restrict__ outV) {
  int bq = blockIdx.x;
  __shared__ float red[256];
  int t = threadIdx.x;
  float xv[4];
  float mx = -3.0e38f;
#pragma unroll
  for (int u = 0; u < 4; ++u) {
    int n = t + u * 256;
    xv[u] = (n < 1000) ? LM[(size_t)bq * 1000 + n] : -3.0e38f;
    mx = fmaxf(mx, xv[u]);
  }
  red[t] = mx;
  __syncthreads();
  for (int s = 128; s > 0; s >>= 1) {
    if (t < s) red[t] = fmaxf(red[t], red[t + s]);
    __syncthreads();
  }
  mx = red[0];
  __syncthreads();
  float ev[4];
  float es = 0.f;
#pragma unroll
  for (int u = 0; u < 4; ++u) {
    int n = t + u * 256;
    ev[u] = (n < 1000) ? __expf(xv[u] - mx) : 0.f;
    es += ev[u];
  }
  red[t] = es;
  __syncthreads();
  for (int s = 128; s > 0; s >>= 1) {
    if (t < s) red[t] += red[t + s];
    __syncthreads();
  }
  float inv = 1.f / red[0];
#pragma unroll
  for (int u = 0; u < 4; ++u) {
    int n = t + u * 256;
    if (n < 1000) {
      float p = ev[u] * inv;
      float j = p * (1.f - p);
      outM[(size_t)bq * 1000 + n] = p;
      outV[(size_t)bq * 1000 + n] = j * j * LV[(size_t)bq * 1000 + n];
    }
  }
}

// ---------------------------------------------------------------------------
extern "C" void kernel_launch(void* const* d_in, const int* in_sizes, int n_in,
                              void* d_out, int out_size, void* d_ws,
                              size_t ws_size, hipStream_t stream) {
  (void)in_sizes; (void)n_in; (void)out_size; (void)ws_size;
  const float* x      = (const float*)d_in[0];
  const float* emb_W  = (const float*)d_in[1];
  const float* emb_b  = (const float*)d_in[2];
  const float* n1_g   = (const float*)d_in[3];
  const float* n1_b   = (const float*)d_in[4];
  const float* Wq     = (const float*)d_in[5];
  const float* Wk     = (const float*)d_in[6];
  const float* Wv     = (const float*)d_in[7];
  const float* n2_g   = (const float*)d_in[8];
  const float* n2_b   = (const float*)d_in[9];
  const float* fc_W   = (const float*)d_in[10];
  const float* fc_b   = (const float*)d_in[11];
  const float* clas_W = (const float*)d_in[12];
  const float* clas_b = (const float*)d_in[13];
  const float* out_W  = (const float*)d_in[14];
  const float* out_b  = (const float*)d_in[15];

  float* ws = (float*)d_ws;
  size_t off = 0;
  auto alloc = [&](size_t n) { float* p = ws + off; off += n; return p; };
  const size_t MK = (size_t)M_ * K_;
  float* hA  = alloc(MK); float* vA  = alloc(MK);
  float* hB  = alloc(MK); float* vB  = alloc(MK);
  float* Qb  = alloc(MK); float* VQb = alloc(MK);
  float* Kb  = alloc(MK); float* VKb = alloc(MK);
  float* Vb  = alloc(MK); float* VVb = alloc(MK);
  const size_t SC = (size_t)B_ * H_ * S_ * S_;
  float* sc  = alloc(SC); float* vsc = alloc(SC);
  float* rs  = alloc(M_);
  float* hp  = alloc((size_t)B_ * K_); float* vp = alloc((size_t)B_ * K_);
  float* rsb = alloc(B_);
  float* c1  = alloc((size_t)B_ * 512); float* v1 = alloc((size_t)B_ * 512);
  float* rs2 = alloc(B_);
  float* lm  = alloc((size_t)B_ * 1000); float* lv = alloc((size_t)B_ * 1000);

  patch_embed_k<<<dim3(M_, K_ / 256), 256, 0, stream>>>(x, emb_W, emb_b, hA, vA);

  float* hc = hA; float* vc = vA; float* ho = hB; float* vo = vB;
  dim3 gGemm(K_ / GTN, (M_ + GTM - 1) / GTM);
  dim3 gScores((S_ + 15) / 16, (S_ + 15) / 16, B_ * H_);
  dim3 gOut(DH_ / 16, (S_ + 15) / 16, B_ * H_);

  for (int i = 0; i < NL_; ++i) {
    if (i > 0)
      layernorm_vdp_k<<<M_, 256, 0, stream>>>(hc, vc, n1_g + (size_t)(i - 1) * K_,
                                              n1_b + (size_t)(i - 1) * K_);
    rowsum_k<<<M_, 256, 0, stream>>>(hc, vc, rs, K_);
    gemm_vdp_k<<<gGemm, 256, 0, stream>>>(hc, vc, Wq + (size_t)i * K_ * K_,
                                          nullptr, rs, nullptr, nullptr, Qb, VQb,
                                          M_, K_, K_, 0);
    gemm_vdp_k<<<gGemm, 256, 0, stream>>>(hc, vc, Wk + (size_t)i * K_ * K_,
                                          nullptr, rs, nullptr, nullptr, Kb, VKb,
                                          M_, K_, K_, 0);
    gemm_vdp_k<<<gGemm, 256, 0, stream>>>(hc, vc, Wv + (size_t)i * K_ * K_,
                                          nullptr, rs, nullptr, nullptr, Vb, VVb,
                                          M_, K_, K_, 0);
    attn_scores_k<<<gScores, 32, 0, stream>>>(Qb, VQb, Kb, VKb, sc, vsc);
    softmax_vdp_k<<<B_ * H_ * S_, 256, 0, stream>>>(sc, vsc);
    attn_out_k<<<gOut, 32, 0, stream>>>(sc, vsc, Vb, VVb, hc, vc, ho, vo);
    { float* t = hc; hc = ho; ho = t; t = vc; vc = vo; vo = t; }
    layernorm_vdp_k<<<M_, 256, 0, stream>>>(hc, vc, n2_g + (size_t)i * K_,
                                            n2_b + (size_t)i * K_);
    rowsum_k<<<M_, 256, 0, stream>>>(hc, vc, rs, K_);
    gemm_vdp_k<<<gGemm, 256, 0, stream>>>(hc, vc, fc_W + (size_t)i * K_ * K_,
                                          fc_b + (size_t)i * K_, rs, hc, vc, ho,
                                          vo, M_, K_, K_, 1 | 2 | 4);
    { float* t = hc; hc = ho; ho = t; t = vc; vc = vo; vo = t; }
  }

  pool_k<<<dim3(B_, K_ / 256), 256, 0, stream>>>(hc, vc, hp, vp);
  rowsum_k<<<B_, 256, 0, stream>>>(hp, vp, rsb, K_);
  head_linear_k<<<dim3(B_, 2), 256, 0, stream>>>(hp, vp, clas_W, clas_b, rsb, c1,
                                                 v1, 512, K_, 1);
  rowsum_k<<<B_, 256, 0, stream>>>(c1, v1, rs2, 512);
  head_linear_k<<<dim3(B_, 4), 256, 0, stream>>>(c1, v1, out_W, out_b, rs2, lm,
                                                 lv, 1000, 512, 0);
  softmax_out_k<<<B_, 256, 0, stream>>>(lm, lv, (float*)d_out,
                                        (float*)d_out + (size_t)B_ * 1000);
}